// C_BatchNorm2d_14843406975465
// MI455X (gfx1250) — compile-verified
//
#include <hip/hip_runtime.h>

// Problem constants (B=32, C=64, H=128, W=128, 2 components)
#define BB   32
#define CC   64
#define HW   16384              // H*W element-pairs per (b,c) slab
#define SLAB_F4 (HW * 2 / 4)    // 8192 float4 per slab
#define NPC  (BB * HW)          // elements per channel for stats

typedef __attribute__((ext_vector_type(2))) float v2f;
typedef __attribute__((ext_vector_type(4))) float v4f;
typedef __attribute__((ext_vector_type(8))) float v8f;

// ---------------------------------------------------------------------------
// Pass 1: per-(b,c)-slab partial stats. Deterministic (no float atomics).
// Each block streams a contiguous 128KB slab with b128 loads and reduces
// 5 moments: sum_x, sum_y, sum_xx, sum_xy, sum_yy.
// ---------------------------------------------------------------------------
__global__ __launch_bounds__(256) void stats_kernel(const float* __restrict__ z,
                                                    float* __restrict__ partial) {
  const int tid = threadIdx.x;
  const v4f* p = (const v4f*)z + (size_t)blockIdx.x * SLAB_F4;

  float sx = 0.f, sy = 0.f, sxx = 0.f, sxy = 0.f, syy = 0.f;
#pragma unroll 4
  for (int i = 0; i < 32; ++i) {
    v4f v = p[i * 256 + tid];
    sx += v.x + v.z;
    sy += v.y + v.w;
    sxx = fmaf(v.x, v.x, fmaf(v.z, v.z, sxx));
    sxy = fmaf(v.x, v.y, fmaf(v.z, v.w, sxy));
    syy = fmaf(v.y, v.y, fmaf(v.w, v.w, syy));
  }

  // wave32 tree reduction
#pragma unroll
  for (int off = 16; off > 0; off >>= 1) {
    sx  += __shfl_down(sx,  off);
    sy  += __shfl_down(sy,  off);
    sxx += __shfl_down(sxx, off);
    sxy += __shfl_down(sxy, off);
    syy += __shfl_down(syy, off);
  }

  __shared__ float red[8][5];
  const int wave = tid >> 5, lane = tid & 31;
  if (lane == 0) {
    red[wave][0] = sx; red[wave][1] = sy; red[wave][2] = sxx;
    red[wave][3] = sxy; red[wave][4] = syy;
  }
  __syncthreads();
  if (tid < 5) {
    float t = 0.f;
#pragma unroll
    for (int w = 0; w < 8; ++w) t += red[w][tid];
    partial[(size_t)blockIdx.x * 5 + tid] = t;  // blockIdx.x == b*CC + c
  }
}

// ---------------------------------------------------------------------------
// Pass 2: one wave per channel. Reduce the 32 batch-partials per stat with
// V_WMMA_F32_16X16X4_F32 accumulations (A = ones  =>  D[m][n] = sum_k B[k][n],
// invariant to the K<->lane packing), then compute the analytic 2x2 inverse
// sqrt and fuse gamma/beta into one per-channel affine {M (2x2), v (2)}.
// ---------------------------------------------------------------------------
__global__ __launch_bounds__(32) void finalize_kernel(const float* __restrict__ partial,
                                                      const float* __restrict__ gamma,
                                                      const float* __restrict__ beta,
                                                      float* __restrict__ params) {
  const int c    = blockIdx.x;
  const int lane = threadIdx.x;
  const int s    = lane & 15;   // column n = stat index (0..4 valid)
  const int kh   = lane >> 4;   // which K-row half this lane supplies

  v8f acc = {0.f, 0.f, 0.f, 0.f, 0.f, 0.f, 0.f, 0.f};
  v2f ones; ones[0] = 1.f; ones[1] = 1.f;  // A = 16x4 all-ones

#pragma unroll
  for (int j = 0; j < 8; ++j) {            // 8 x (K=4) covers the 32 batch partials
    const int b0 = j * 4 + kh;
    const int b1 = j * 4 + 2 + kh;
    v2f bm;
    bm[0] = (s < 5) ? partial[((size_t)b0 * CC + c) * 5 + s] : 0.f;
    bm[1] = (s < 5) ? partial[((size_t)b1 * CC + c) * 5 + s] : 0.f;
    acc = __builtin_amdgcn_wmma_f32_16x16x4_f32(
        /*neg_a=*/false, ones, /*neg_b=*/false, bm,
        /*c_mod=*/(short)0, acc, /*reuse_a=*/false, /*reuse_b=*/false);
  }

  // D row M=0 lives in acc[0], N = lane (lanes 0..15): totals per stat
  const float t = acc[0];
  const float S0 = __shfl(t, 0), S1 = __shfl(t, 1), S2 = __shfl(t, 2),
              S3 = __shfl(t, 3), S4 = __shfl(t, 4);

  const float invN = 1.0f / (float)NPC;
  const float mu0 = S0 * invN, mu1 = S1 * invN;
  const float a  = S2 * invN - mu0 * mu0 + 1e-5f;
  const float bc = S3 * invN - mu0 * mu1;
  const float d  = S4 * invN - mu1 * mu1 + 1e-5f;

  // analytic 2x2 inverse matrix square root (matches reference _inv_sqrt_22)
  const float sdet = sqrtf(a * d - bc * bc);
  const float tr   = sqrtf(a + d + 2.0f * sdet);
  const float sa = (a + sdet) / tr, sb = bc / tr, sc = bc / tr, sd = (d + sdet) / tr;
  const float det = sa * sd - sb * sc;
  const float w00 = sd / det, w01 = -sb / det, w10 = -sc / det, w11 = sa / det;

  const float g00 = gamma[0], g01 = gamma[1], g10 = gamma[2], g11 = gamma[3];
  const float M00 = g00 * w00 + g01 * w10, M01 = g00 * w01 + g01 * w11;
  const float M10 = g10 * w00 + g11 * w10, M11 = g10 * w01 + g11 * w11;
  const float v0 = beta[0] - (M00 * mu0 + M01 * mu1);
  const float v1 = beta[1] - (M10 * mu0 + M11 * mu1);

  if (lane == 0) {
    float* pr = params + c * 8;
    pr[0] = M00; pr[1] = M01; pr[2] = M10; pr[3] = M11;
    pr[4] = v0;  pr[5] = v1;  pr[6] = 0.f; pr[7] = 0.f;
  }
}

// ---------------------------------------------------------------------------
// Pass 3: streaming apply. NT b128 loads (last use of z), per-pair 2x2 affine,
// non-temporal b128 stores (output is write-once).
// ---------------------------------------------------------------------------
__global__ __launch_bounds__(256) void apply_kernel(const float* __restrict__ z,
                                                    const float* __restrict__ params,
                                                    float* __restrict__ out) {
  const int blk = blockIdx.x;           // 8 blocks per (b,c) slab
  const int c   = (blk >> 3) & (CC - 1);

  const v4f* pp = (const v4f*)params;
  const v4f m  = pp[c * 2];             // M00 M01 M10 M11
  const v4f vb = pp[c * 2 + 1];         // v0 v1 - -

  const size_t base = (size_t)blk * 1024;
  const v4f* in4 = (const v4f*)z;
  v4f* out4 = (v4f*)out;

#pragma unroll
  for (int i = 0; i < 4; ++i) {
    const size_t idx = base + (size_t)i * 256 + threadIdx.x;
    v4f v = __builtin_nontemporal_load(&in4[idx]);
    v4f o;
    o.x = fmaf(m.x, v.x, fmaf(m.y, v.y, vb.x));
    o.y = fmaf(m.z, v.x, fmaf(m.w, v.y, vb.y));
    o.z = fmaf(m.x, v.z, fmaf(m.y, v.w, vb.x));
    o.w = fmaf(m.z, v.z, fmaf(m.w, v.w, vb.y));
    __builtin_nontemporal_store(o, &out4[idx]);
  }
}

// ---------------------------------------------------------------------------
extern "C" void kernel_launch(void* const* d_in, const int* in_sizes, int n_in,
                              void* d_out, int out_size, void* d_ws, size_t ws_size,
                              hipStream_t stream) {
  const float* z     = (const float*)d_in[0];
  const float* gamma = (const float*)d_in[1];
  const float* beta  = (const float*)d_in[2];
  float* out = (float*)d_out;
  float* ws  = (float*)d_ws;

  float* partial = ws;            // 2048 * 5 floats
  float* params  = ws + 10240;    // 64 * 8 floats (16B aligned offset)

  stats_kernel<<<BB * CC, 256, 0, stream>>>(z, partial);          // 2048 blocks
  finalize_kernel<<<CC, 32, 0, stream>>>(partial, gamma, beta, params);
  apply_kernel<<<BB * CC * 8, 256, 0, stream>>>(z, params, out);  // 16384 blocks
}